// GConvMultiScale_23029614641203
// MI455X (gfx1250) — compile-verified
//
#include <hip/hip_runtime.h>
#include <hip/hip_bf16.h>

// ---------------------------------------------------------------------------
// Multi-scale GCN (GConv) for MI455X / gfx1250.
//   For each scale t in 0..2:
//     deg[i]  = sum_{e: col=i} w[e] + 1 (self loop)
//     dinv    = rsqrt(deg)
//     layer(z): h = z @ W (f32 WMMA 16x16x4)
//               agg[c] += h[r] * dinv[r]*w*dinv[c]   (edge scatter, L2 atomics)
//               z' = PReLU(agg + h*dinv^2 + b)       (self loop folded in)
//     out += coeff[t] * z2
// ---------------------------------------------------------------------------

typedef __attribute__((ext_vector_type(2))) float v2f;
typedef __attribute__((ext_vector_type(8))) float v8f;

#define D_FEAT 128

// ---------------- zero fill ----------------
__global__ void zero_kernel(float* __restrict__ p, int n) {
  int i = blockIdx.x * blockDim.x + threadIdx.x;
  if (i < n) p[i] = 0.0f;
}

// ---------------- degree accumulation ----------------
__global__ void deg_kernel(const int* __restrict__ col,
                           const float* __restrict__ w,
                           float* __restrict__ deg, int E) {
  int e = blockIdx.x * blockDim.x + threadIdx.x;
  if (e < E) atomicAdd(&deg[col[e]], w[e]);
}

// ---------------- dinv = rsqrt(deg + 1)  (self-loop weight folded) ---------
__global__ void dinv_kernel(float* __restrict__ deg, int n) {
  int i = blockIdx.x * blockDim.x + threadIdx.x;
  if (i < n) {
    float d = deg[i] + 1.0f;  // self loop contributes weight 1.0; d >= 1 > 0
    deg[i] = rsqrtf(d);
  }
}

// ---------------- GEMM: H[n x 128] = Z[n x 128] * W[128 x 128] -------------
// One wave32 per 16x16 output tile, 32 x v_wmma_f32_16x16x4_f32 along K=128.
// 8 waves per 256-thread block.  nRows % 16 == 0 (50000), so EXEC is all-1s
// for every wave that enters the WMMA loop (WMMA requirement).
__global__ __launch_bounds__(256) void gemm_wmma_kernel(
    const float* __restrict__ Z, const float* __restrict__ W,
    float* __restrict__ H, int nRows) {
  const int lane = threadIdx.x;                 // 0..31
  const int wave = threadIdx.y;                 // 0..7
  const int tile = blockIdx.x * 8 + wave;
  const int nTiles = (nRows / 16) * (D_FEAT / 16);
  if (tile >= nTiles) return;                   // wave-uniform branch

  const int m0 = (tile / (D_FEAT / 16)) * 16;
  const int n0 = (tile % (D_FEAT / 16)) * 16;
  const int l15   = lane & 15;
  const int khalf = (lane >> 4) << 1;           // 0 (lanes 0-15) or 2 (16-31)

  const float* zrow = Z + (size_t)(m0 + l15) * D_FEAT;
  v8f acc = {};

#pragma unroll
  for (int k0 = 0; k0 < D_FEAT; k0 += 4) {
    const int kb = k0 + khalf;
    // A fragment (16x4 f32): lane<16 holds row m0+l15, K = k0,k0+1
    //                        lane>=16 holds row m0+l15, K = k0+2,k0+3
    v2f a;
    a.x = zrow[kb];
    a.y = zrow[kb + 1];
    // B fragment (4x16 f32): lane<16 holds col n0+l15, K = k0,k0+1 ; etc.
    v2f b;
    b.x = W[(size_t)kb * D_FEAT + n0 + l15];
    b.y = W[(size_t)(kb + 1) * D_FEAT + n0 + l15];
    acc = __builtin_amdgcn_wmma_f32_16x16x4_f32(
        /*neg_a=*/false, a, /*neg_b=*/false, b,
        /*c_mod=*/(short)0, acc, /*reuse_a=*/false, /*reuse_b=*/false);
  }

  // C/D layout: VGPR r -> lanes 0-15: M=m0+r, lanes 16-31: M=m0+8+r ; N=n0+l15
  const int mbase = m0 + ((lane >> 4) << 3);
  float* hp = H + (size_t)mbase * D_FEAT + n0 + l15;
#pragma unroll
  for (int r = 0; r < 8; ++r) hp[(size_t)r * D_FEAT] = acc[r];
}

// ---------------- edge scatter: agg[c] += h[r] * norm ----------------------
// One wave per edge; each lane moves a float4 of the 128-wide feature row.
// h / agg are L2-resident (25.6 MB each vs 192 MB L2), so the gather hits L2
// and the atomics are L2-side f32 adds.
__global__ __launch_bounds__(256) void aggregate_kernel(
    const int* __restrict__ row, const int* __restrict__ col,
    const float* __restrict__ w, const float* __restrict__ dinv,
    const float* __restrict__ H, float* __restrict__ AGG, int E) {
  const int lane = threadIdx.x;                 // 0..31
  const int e = blockIdx.x * 8 + threadIdx.y;   // wave-uniform edge id
  if (e >= E) return;
  const int r = row[e];
  const int c = col[e];
  const float nrm = dinv[r] * w[e] * dinv[c];

  const float4 v = ((const float4*)(H + (size_t)r * D_FEAT))[lane];
  float* ap = AGG + (size_t)c * D_FEAT + lane * 4;
  atomicAdd(ap + 0, v.x * nrm);
  atomicAdd(ap + 1, v.y * nrm);
  atomicAdd(ap + 2, v.z * nrm);
  atomicAdd(ap + 3, v.w * nrm);
}

// ---------------- epilogue: self-loop + bias + PReLU (in place on AGG) -----
__global__ void finish_kernel(float* __restrict__ Z /* = AGG in/out */,
                              const float* __restrict__ H,
                              const float* __restrict__ dinv,
                              const float* __restrict__ bias,
                              const float* __restrict__ pw, int total) {
  int i = blockIdx.x * blockDim.x + threadIdx.x;
  if (i >= total) return;
  const int f = i & (D_FEAT - 1);
  const int node = i >> 7;
  const float d = dinv[node];
  const float v = Z[i] + H[i] * d * d + bias[f];   // self-loop norm = dinv^2
  Z[i] = fmaxf(v, 0.0f) + pw[f] * fminf(v, 0.0f);  // per-channel PReLU
}

// ---------------- out += coeff[t] * z2 ----------------
__global__ void accum_kernel(const float* __restrict__ Z,
                             const float* __restrict__ coeff, int t,
                             float* __restrict__ out, int total) {
  int i = blockIdx.x * blockDim.x + threadIdx.x;
  if (i >= total) return;
  out[i] += coeff[t] * Z[i];
}

// ---------------------------------------------------------------------------
extern "C" void kernel_launch(void* const* d_in, const int* in_sizes, int n_in,
                              void* d_out, int out_size, void* d_ws, size_t ws_size,
                              hipStream_t stream) {
  const float* x       = (const float*)d_in[0];  // [N,128]
  const int*   eidx    = (const int*)  d_in[1];  // [3,2,E]
  const float* ew      = (const float*)d_in[2];  // [3,E]
  const float* coeff   = (const float*)d_in[3];  // [3]
  const float* W1      = (const float*)d_in[4];  // [128,128]
  const float* b1      = (const float*)d_in[5];  // [128]
  const float* W2      = (const float*)d_in[6];  // [128,128]
  const float* b2      = (const float*)d_in[7];  // [128]
  const float* prelu_w = (const float*)d_in[8];  // [128]

  const int N = in_sizes[0] / D_FEAT;            // 50000
  const int E = in_sizes[2] / 3;                 // 800000
  const int total = N * D_FEAT;

  // Workspace layout (floats): [dinv: N (padded)] [bufA: N*128] [bufB: N*128]
  float* ws   = (float*)d_ws;
  float* dinv = ws;
  float* bufA = ws + (((size_t)N + 31) & ~(size_t)31);
  float* bufB = bufA + (size_t)N * D_FEAT;

  const int ETHREADS = 256;
  const dim3 waveBlock(32, 8);                               // 8 wave32 / block
  const int gemmBlocks = ((N / 16) * (D_FEAT / 16) + 7) / 8; // wave per tile
  const int edgeBlocks = (E + 7) / 8;                        // wave per edge
  const int elemBlocks = (total + ETHREADS - 1) / ETHREADS;
  const int nodeBlocks = (N + ETHREADS - 1) / ETHREADS;

  // out = 0
  zero_kernel<<<elemBlocks, ETHREADS, 0, stream>>>((float*)d_out, total);

  for (int t = 0; t < 3; ++t) {
    const int* row = eidx + (size_t)t * 2 * E;
    const int* col = row + E;
    const float* w = ew + (size_t)t * E;

    // degree -> dinv
    zero_kernel<<<nodeBlocks, ETHREADS, 0, stream>>>(dinv, N);
    deg_kernel<<<(E + ETHREADS - 1) / ETHREADS, ETHREADS, 0, stream>>>(col, w, dinv, E);
    dinv_kernel<<<nodeBlocks, ETHREADS, 0, stream>>>(dinv, N);

    // ---- layer 1: z = x ----
    gemm_wmma_kernel<<<gemmBlocks, waveBlock, 0, stream>>>(x, W1, bufA, N);
    zero_kernel<<<elemBlocks, ETHREADS, 0, stream>>>(bufB, total);
    aggregate_kernel<<<edgeBlocks, waveBlock, 0, stream>>>(row, col, w, dinv, bufA, bufB, E);
    finish_kernel<<<elemBlocks, ETHREADS, 0, stream>>>(bufB, bufA, dinv, b1, prelu_w, total);

    // ---- layer 2: z = bufB ----
    gemm_wmma_kernel<<<gemmBlocks, waveBlock, 0, stream>>>(bufB, W2, bufA, N);
    zero_kernel<<<elemBlocks, ETHREADS, 0, stream>>>(bufB, total);
    aggregate_kernel<<<edgeBlocks, waveBlock, 0, stream>>>(row, col, w, dinv, bufA, bufB, E);
    finish_kernel<<<elemBlocks, ETHREADS, 0, stream>>>(bufB, bufA, dinv, b2, prelu_w, total);

    // ---- out += coeff[t] * z2 ----
    accum_kernel<<<elemBlocks, ETHREADS, 0, stream>>>(bufB, coeff, t, (float*)d_out, total);
  }
}